// GCNWithSkipConnections_30949534335550
// MI455X (gfx1250) — compile-verified
//
#include <hip/hip_runtime.h>

typedef __attribute__((ext_vector_type(2))) float v2f;
typedef __attribute__((ext_vector_type(8))) float v8f;
typedef __attribute__((ext_vector_type(4))) unsigned int tdm_g0_t;
typedef __attribute__((ext_vector_type(8))) int tdm_g1_t;
typedef __attribute__((ext_vector_type(4))) int tdm_g23_t;
typedef __attribute__((ext_vector_type(8))) int tdm_g4_t;

#define FEAT 128

// ---------------- utility kernels ----------------

__global__ __launch_bounds__(256) void zero_f32(float* __restrict__ p, int n) {
  int i = blockIdx.x * 256 + threadIdx.x;
  if (i < n) p[i] = 0.0f;
}

__global__ __launch_bounds__(256) void deg_count(const int* __restrict__ dst,
                                                 float* __restrict__ deg, int E) {
  int e = blockIdx.x * 256 + threadIdx.x;
  if (e < E) atomicAdd(&deg[dst[e]], 1.0f);
}

__global__ __launch_bounds__(256) void make_dinv(float* __restrict__ deg_dinv, int N) {
  int n = blockIdx.x * 256 + threadIdx.x;
  if (n < N) deg_dinv[n] = rsqrtf(deg_dinv[n] + 1.0f);
}

// ---------------- fp32 WMMA GEMM: C(N x 128) = A(N x 128) * W(128 x 128) ----------------
// 256 threads = 8 wave32; each wave owns a 16-row strip and all 8 column tiles.
// W (64 KB) staged into LDS by the Tensor Data Mover (TDM) when available.
__global__ __launch_bounds__(256) void gemm_xw(const float* __restrict__ A,
                                               const float* __restrict__ W,
                                               float* __restrict__ C, int nRows) {
  __shared__ float wlds[FEAT * FEAT];
  const int tid = threadIdx.x;

#if defined(__gfx1250__) && __has_builtin(__builtin_amdgcn_tensor_load_to_lds) && \
    __has_builtin(__builtin_amdgcn_s_wait_tensorcnt)
  if (tid < 32) {
    // D# group 0: count=1 | lds_addr | global_addr[56:0] | type=2
    unsigned lds_off = (unsigned)(unsigned long long)&wlds[0];  // low 32b = LDS byte addr
    unsigned long long ga = (unsigned long long)W;
    tdm_g0_t g0 = {1u, lds_off, (unsigned)ga,
                   (unsigned)((ga >> 32) & 0x01FFFFFFu) | 0x80000000u};
    // D# group 1: data_size=4B; tensor_dim0/1 = 128; tile_dim0/1 = 128; dim0_stride = 128
    tdm_g1_t g1 = {(int)0x00020000, (int)(128u << 16), (int)(128u << 16),
                   (int)(128u << 16), 128, 128, 0, 0};
    tdm_g23_t gz = {0, 0, 0, 0};
#if __clang_major__ >= 23
    tdm_g4_t g4 = {0, 0, 0, 0, 0, 0, 0, 0};
    __builtin_amdgcn_tensor_load_to_lds(g0, g1, gz, gz, g4, 0);
#else
    __builtin_amdgcn_tensor_load_to_lds(g0, g1, gz, gz, 0);
#endif
    __builtin_amdgcn_s_wait_tensorcnt(0);
  }
#else
  {
    const float4* W4 = (const float4*)W;
    float4* L4 = (float4*)wlds;
#pragma unroll
    for (int i = 0; i < 16; ++i) L4[i * 256 + tid] = W4[i * 256 + tid];
  }
#endif
  __syncthreads();

  const int wave = tid >> 5;
  const int lane = tid & 31;
  const int half = lane >> 4;  // 0: lanes 0-15, 1: lanes 16-31
  const int l15  = lane & 15;
  const int mbase = blockIdx.x * 128 + wave * 16;

  int arow = mbase + l15;                  // A-frag row for this lane
  if (arow >= nRows) arow = nRows - 1;     // clamp (keeps EXEC full; stores are guarded)
  const float* __restrict__ Arow = A + (size_t)arow * FEAT;

  v8f zero = {};
  v8f acc[8];
#pragma unroll
  for (int t = 0; t < 8; ++t) acc[t] = zero;

  for (int k = 0; k < FEAT; k += 4) {
    // A 16x4 fp32 frag: VGPR0/1 hold K = k+2*half, k+2*half+1 (contiguous pair -> b64 load)
    const int ka = k + 2 * half;
    v2f a;
    a.x = Arow[ka + 0];
    a.y = Arow[ka + 1];
#pragma unroll
    for (int t = 0; t < 8; ++t) {
      const int nc = t * 16 + l15;
      v2f b;  // B 4x16 frag: rows K=ka, ka+1 striped across lanes
      b.x = wlds[(ka + 0) * FEAT + nc];
      b.y = wlds[(ka + 1) * FEAT + nc];
      acc[t] = __builtin_amdgcn_wmma_f32_16x16x4_f32(
          /*neg_a=*/false, a, /*neg_b=*/false, b,
          /*c_mod=*/(short)0, acc[t], /*reuse_a=*/false, /*reuse_b=*/false);
    }
  }

  // C/D layout: VGPR i -> row mbase + i + 8*half, col = tile*16 + (lane&15)
  float* __restrict__ Crow = C + (size_t)(mbase + 8 * half) * FEAT + l15;
  if (mbase + 16 <= nRows) {  // wave-uniform fast path: immediate-offset stores
#pragma unroll
    for (int t = 0; t < 8; ++t)
#pragma unroll
      for (int i = 0; i < 8; ++i) Crow[i * FEAT + t * 16] = acc[t][i];
  } else {
#pragma unroll
    for (int t = 0; t < 8; ++t)
#pragma unroll
      for (int i = 0; i < 8; ++i)
        if (mbase + 8 * half + i < nRows) Crow[i * FEAT + t * 16] = acc[t][i];
  }
}

// ---------------- edge scatter: agg[dst] += tmp[src] * dinv[src]*dinv[dst] ----------------
__global__ __launch_bounds__(128) void edge_scatter(const float* __restrict__ tmp,
                                                    const int* __restrict__ src,
                                                    const int* __restrict__ dst,
                                                    const float* __restrict__ dinv,
                                                    float* __restrict__ agg, int E) {
  const int f = threadIdx.x;
  const int e0 = blockIdx.x * 8;
#pragma unroll
  for (int i = 0; i < 8; ++i) {
    int e = e0 + i;
    if (e >= E) return;
    int s = src[e];  // uniform per block-iteration -> scalarized loads
    int d = dst[e];
    float w = dinv[s] * dinv[d];
    atomicAdd(&agg[(size_t)d * FEAT + f], tmp[(size_t)s * FEAT + f] * w);
  }
}

// ---------------- self_init: agg = tmp*dinv^2 + b (replaces zeroing; runs pre-scatter) ----
__global__ __launch_bounds__(256) void self_init(float* __restrict__ agg,
                                                 const float* __restrict__ tmp,
                                                 const float* __restrict__ dinv,
                                                 const float* __restrict__ bias, int N) {
  int idx = blockIdx.x * 256 + threadIdx.x;
  if (idx >= N * FEAT) return;
  int node = idx >> 7;
  int f = idx & (FEAT - 1);
  float dn = dinv[node];
  agg[idx] = tmp[idx] * dn * dn + bias[f];
}

// ---------------- finalize: h_next = relu(agg) (+ skip), in place in agg ----------------
__global__ __launch_bounds__(256) void finalize(float* __restrict__ agg,
                                                const float* __restrict__ hprev,
                                                int useSkip, int N) {
  int idx = blockIdx.x * 256 + threadIdx.x;
  if (idx >= N * FEAT) return;
  float v = fmaxf(agg[idx], 0.0f);
  if (useSkip) v += hprev[idx];
  agg[idx] = v;
}

// ---------------- pooling ----------------

__global__ __launch_bounds__(256) void count_nodes(const int* __restrict__ batch,
                                                   float* __restrict__ counts, int N) {
  int n = blockIdx.x * 256 + threadIdx.x;
  if (n < N) atomicAdd(&counts[batch[n]], 1.0f);
}

__global__ __launch_bounds__(128) void pool_scatter(const float* __restrict__ h,
                                                    const int* __restrict__ batch,
                                                    float* __restrict__ pooled) {
  int n = blockIdx.x;
  int f = threadIdx.x;
  int g = batch[n];  // uniform per block -> scalarized
  atomicAdd(&pooled[(size_t)g * FEAT + f], h[(size_t)n * FEAT + f]);
}

__global__ __launch_bounds__(128) void pool_div(float* __restrict__ pooled,
                                                const float* __restrict__ counts) {
  int g = blockIdx.x;
  int f = threadIdx.x;
  pooled[(size_t)g * FEAT + f] /= fmaxf(counts[g], 1.0f);
}

// ---------------- MLP head (tiny: 256x128 @ 128x128, 256x128 @ 128x1) ----------------

__global__ __launch_bounds__(128) void mlp1(const float* __restrict__ pooled,
                                            const float* __restrict__ lw1,
                                            const float* __restrict__ lb1,
                                            float* __restrict__ h2) {
  int g = blockIdx.x;
  int j = threadIdx.x;
  __shared__ float pg[FEAT];
  pg[j] = pooled[(size_t)g * FEAT + j];
  __syncthreads();
  float acc = lb1[j];
#pragma unroll 4
  for (int k = 0; k < FEAT; ++k) acc += pg[k] * lw1[k * FEAT + j];
  h2[(size_t)g * FEAT + j] = fmaxf(acc, 0.0f);
}

__global__ __launch_bounds__(128) void mlp2(const float* __restrict__ h2,
                                            const float* __restrict__ lw2,
                                            const float* __restrict__ lb2,
                                            float* __restrict__ out) {
  int g = blockIdx.x;
  int k = threadIdx.x;
  __shared__ float red[FEAT];
  red[k] = h2[(size_t)g * FEAT + k] * lw2[k];
  __syncthreads();
  for (int s = 64; s > 0; s >>= 1) {
    if (k < s) red[k] += red[k + s];
    __syncthreads();
  }
  if (k == 0) out[g] = red[0] + lb2[0];
}

// ---------------- launcher ----------------

extern "C" void kernel_launch(void* const* d_in, const int* in_sizes, int n_in,
                              void* d_out, int out_size, void* d_ws, size_t ws_size,
                              hipStream_t stream) {
  const float* x    = (const float*)d_in[0];
  const int*   eidx = (const int*)d_in[1];
  const int*   batch= (const int*)d_in[2];
  const float* W1   = (const float*)d_in[3];
  const float* b1   = (const float*)d_in[4];
  const float* Ws   = (const float*)d_in[5];
  const float* bs   = (const float*)d_in[6];
  const float* lw1  = (const float*)d_in[7];
  const float* lb1  = (const float*)d_in[8];
  const float* lw2  = (const float*)d_in[9];
  const float* lb2  = (const float*)d_in[10];
  float* out = (float*)d_out;

  const int N = in_sizes[0] / FEAT;   // 40000
  const int E = in_sizes[1] / 2;      // 640000
  const int G = out_size;             // 256 (O == 1)
  const int* src = eidx;
  const int* dst = eidx + E;

  float* ws = (float*)d_ws;
  size_t o = 0;
  auto alloc = [&](size_t nf) {
    float* p = ws + o;
    o += (nf + 255) & ~(size_t)255;
    return p;
  };
  float* dinv   = alloc((size_t)N);            // deg, then dinv in place
  float* counts = alloc((size_t)G);
  float* pooled = alloc((size_t)G * FEAT);
  float* h2     = alloc((size_t)G * FEAT);
  float* P0     = alloc((size_t)N * FEAT);
  float* P1     = alloc((size_t)N * FEAT);
  float* P2     = alloc((size_t)N * FEAT);
  float* P[3] = {P0, P1, P2};

  // degree + symmetric normalization factors
  zero_f32<<<(N + 255) / 256, 256, 0, stream>>>(dinv, N);
  deg_count<<<(E + 255) / 256, 256, 0, stream>>>(dst, dinv, E);
  make_dinv<<<(N + 255) / 256, 256, 0, stream>>>(dinv, N);

  // buffer rotation across 4 layers: (tmp, agg) indices; cur starts at x
  const int tmpIdx[4] = {0, 0, 1, 2};
  const int aggIdx[4] = {1, 2, 0, 1};
  const float* cur = x;
  const int NH = N * FEAT;
  for (int l = 0; l < 4; ++l) {
    const float* Wl = (l == 0) ? W1 : (Ws + (size_t)(l - 1) * FEAT * FEAT);
    const float* bl = (l == 0) ? b1 : (bs + (size_t)(l - 1) * FEAT);
    float* tmp = P[tmpIdx[l]];
    float* agg = P[aggIdx[l]];
    gemm_xw<<<(N + 127) / 128, 256, 0, stream>>>(cur, Wl, tmp, N);
    self_init<<<(NH + 255) / 256, 256, 0, stream>>>(agg, tmp, dinv, bl, N);
    edge_scatter<<<(E + 7) / 8, 128, 0, stream>>>(tmp, src, dst, dinv, agg, E);
    finalize<<<(NH + 255) / 256, 256, 0, stream>>>(agg, cur, (l > 0) ? 1 : 0, N);
    cur = agg;  // layer output lives in agg buffer
  }

  // global mean pool
  zero_f32<<<(G + 255) / 256, 256, 0, stream>>>(counts, G);
  zero_f32<<<(G * FEAT + 255) / 256, 256, 0, stream>>>(pooled, G * FEAT);
  count_nodes<<<(N + 255) / 256, 256, 0, stream>>>(batch, counts, N);
  pool_scatter<<<N, 128, 0, stream>>>(cur, batch, pooled);
  pool_div<<<G, 128, 0, stream>>>(pooled, counts);

  // MLP head
  mlp1<<<G, 128, 0, stream>>>(pooled, lw1, lb1, h2);
  mlp2<<<G, 128, 0, stream>>>(h2, lw2, lb2, out);
}